// TWIRLSConv_6399501271284
// MI455X (gfx1250) — compile-verified
//
#include <hip/hip_runtime.h>

// TWIRLSConv for MI455X (gfx1250, wave32).
//  - Dense MLPs via V_WMMA_F32_16X16X4_F32 (fp32-exact, wave-per-16x16-tile)
//  - Graph propagation: fused nrm-gather + float4 atomic scatter (16 lanes/edge)
//  - LAM=1, ALP=0.5, TAU=0.2, P_RHO=1, no upper threshold

typedef __attribute__((ext_vector_type(2))) float v2f;
typedef __attribute__((ext_vector_type(8))) float v8f;

#define LAM_F 1.0f
#define TAU_F 0.2f

__global__ __launch_bounds__(256)
void zero_f(float* __restrict__ p, long n) {
  long t = (long)blockIdx.x * blockDim.x + threadIdx.x;
  if (t < n) p[t] = 0.0f;
}

// w[e] = 1, deg[dst] += 1
__global__ __launch_bounds__(256)
void init_edges(const int* __restrict__ dst, float* __restrict__ w,
                float* __restrict__ deg, int E) {
  int e = blockIdx.x * blockDim.x + threadIdx.x;
  if (e < E) {
    w[e] = 1.0f;
    atomicAdd(&deg[dst[e]], 1.0f);
  }
}

// dt = lam*deg + (1-lam);  nrm = dt^-1/2; invdt = 1/dt
__global__ __launch_bounds__(256)
void node_coeffs(const float* __restrict__ deg, float* __restrict__ nrm,
                 float* __restrict__ invdt, int N) {
  int i = blockIdx.x * blockDim.x + threadIdx.x;
  if (i < N) {
    float dt = LAM_F * deg[i] + (1.0f - LAM_F);
    nrm[i]   = rsqrtf(dt);
    invdt[i] = 1.0f / dt;
  }
}

// P[dst] += w[e]*nrm[src]*Y[src]   (16 lanes per edge, float4 per lane)
__global__ __launch_bounds__(256)
void edge_scatter(const int* __restrict__ src, const int* __restrict__ dst,
                  const float* __restrict__ w, const float* __restrict__ nrm,
                  const float* __restrict__ Y, float* __restrict__ P, int E) {
  int t = blockIdx.x * blockDim.x + threadIdx.x;
  int e = t >> 4;
  if (e >= E) return;
  int q = (t & 15) << 2;
  int s = src[e], d = dst[e];
  float c = w[e] * nrm[s];
  const float4 y = *(const float4*)(Y + (size_t)s * 64 + q);
  float* pp = P + (size_t)d * 64 + q;
  atomicAdd(pp + 0, c * y.x);
  atomicAdd(pp + 1, c * y.y);
  atomicAdd(pp + 2, c * y.z);
  atomicAdd(pp + 3, c * y.w);
}

// Y = 0.5*Y + 0.5*nrm*P + 0.5*invdt*X   (float4 per thread)
__global__ __launch_bounds__(256)
void node_update(const float* __restrict__ X, const float* __restrict__ P,
                 const float* __restrict__ nrm, const float* __restrict__ invdt,
                 float* __restrict__ Y, int N) {
  int t = blockIdx.x * blockDim.x + threadIdx.x;   // float4 index
  if (t >= N * 16) return;
  int i = t >> 4;
  float4 y = ((const float4*)Y)[t];
  float4 p = ((const float4*)P)[t];
  float4 x = ((const float4*)X)[t];
  float a = 0.5f * nrm[i];
  float b = 0.5f * invdt[i];
  float4 r;
  r.x = 0.5f * y.x + a * p.x + b * x.x;
  r.y = 0.5f * y.y + a * p.y + b * x.y;
  r.z = 0.5f * y.z + a * p.z + b * x.z;
  r.w = 0.5f * y.w + a * p.w + b * x.w;
  ((float4*)Y)[t] = r;
}

// w[e] = 1/max(sqrt(||Y[s]-Y[d]||^2 + 1e-7), tau) + 1e-9;  deg[dst] += w
__global__ __launch_bounds__(256)
void edge_attention(const int* __restrict__ src, const int* __restrict__ dst,
                    const float* __restrict__ Y, float* __restrict__ w,
                    float* __restrict__ deg, int E) {
  int e = blockIdx.x * blockDim.x + threadIdx.x;
  if (e >= E) return;
  int s = src[e], d = dst[e];
  const float4* ys = (const float4*)(Y + (size_t)s * 64);
  const float4* yd = (const float4*)(Y + (size_t)d * 64);
  float acc = 0.0f;
#pragma unroll
  for (int i = 0; i < 16; ++i) {
    float4 a = ys[i];
    float4 b = yd[i];
    float dx = a.x - b.x, dy = a.y - b.y, dz = a.z - b.z, dw = a.w - b.w;
    acc += dx * dx + dy * dy + dz * dz + dw * dw;
  }
  float wv = acc + 1e-7f;        // relu(||.||^2) + eps ; acc >= 0 by construction
  wv = sqrtf(wv);                // pow(w, 1 - 0.5*rho), rho = 1
  wv = fmaxf(wv, TAU_F);
  wv = 1.0f / wv + 1e-9f;
  w[e] = wv;
  atomicAdd(&deg[d], wv);
}

// C[M,64] = act(A[M,K]) @ B[K,64] + bias; optional second destination C2.
// One wave per 16x16 tile; K-loop via V_WMMA_F32_16X16X4_F32.
// A frag (16x4 f32): lane half selects K pair {k,k+1} / {k+2,k+3}, lane&15 = M.
// B frag (4x16 f32): VGPR0 = rows k / k+2, VGPR1 = rows k+1 / k+3; lane&15 = N.
// C/D (16x16 f32): VGPR j -> M = j (+8 upper half), N = lane&15.
template <int K, bool RELU>
__global__ __launch_bounds__(256)
void wmma_gemm64(const float* __restrict__ A, const float* __restrict__ B,
                 const float* __restrict__ bias, float* __restrict__ C,
                 float* __restrict__ C2, int M) {
  int wave = (blockIdx.x * blockDim.x + threadIdx.x) >> 5;
  int lane = threadIdx.x & 31;
  int tiles = (M >> 4) * 4;                 // 4 column tiles (64 output cols)
  if (wave >= tiles) return;                // wave-uniform: EXEC stays all-ones
  int row0 = (wave >> 2) << 4;
  int col0 = (wave & 3) << 4;
  int half = lane >> 4;                     // 0 or 1
  int mr   = lane & 15;

  const float* arow = A + (size_t)(row0 + mr) * K;
  v8f acc = {};
#pragma unroll
  for (int k = 0; k < K; k += 4) {
    int ka = k + half * 2;
    float a0 = arow[ka];
    float a1 = arow[ka + 1];
    if (RELU) { a0 = fmaxf(a0, 0.0f); a1 = fmaxf(a1, 0.0f); }
    v2f a, b;
    a[0] = a0;
    a[1] = a1;
    b[0] = B[(size_t)(ka + 0) * 64 + col0 + mr];
    b[1] = B[(size_t)(ka + 1) * 64 + col0 + mr];
    acc = __builtin_amdgcn_wmma_f32_16x16x4_f32(
        /*neg_a=*/false, a, /*neg_b=*/false, b,
        /*c_mod=*/(short)0, acc, /*reuse_a=*/false, /*reuse_b=*/false);
  }
  float bv = bias[col0 + mr];
#pragma unroll
  for (int j = 0; j < 8; ++j) {
    size_t idx = (size_t)(row0 + half * 8 + j) * 64 + col0 + mr;
    float v = acc[j] + bv;
    C[idx] = v;
    if (C2 != nullptr) C2[idx] = v;
  }
}

extern "C" void kernel_launch(void* const* d_in, const int* in_sizes, int n_in,
                              void* d_out, int out_size, void* d_ws, size_t ws_size,
                              hipStream_t stream) {
  (void)n_in; (void)out_size; (void)ws_size;
  const float* feat  = (const float*)d_in[0];
  const float* W_bef = (const float*)d_in[1];
  const float* b_bef = (const float*)d_in[2];
  const float* W_aft = (const float*)d_in[3];
  const float* b_aft = (const float*)d_in[4];
  const int*   src   = (const int*)d_in[5];
  const int*   dst   = (const int*)d_in[6];

  const int N = in_sizes[0] / 128;   // 100000
  const int E = in_sizes[5];         // 900000

  // workspace carve-out (256B-aligned slices)
  char* p = (char*)d_ws;
  auto take = [&](size_t bytes) -> void* {
    void* r = (void*)p;
    p += (bytes + 255) & ~(size_t)255;
    return r;
  };
  size_t nfb = (size_t)N * 64 * sizeof(float);
  float* X     = (float*)take(nfb);
  float* Y     = (float*)take(nfb);
  float* P     = (float*)take(nfb);
  float* w     = (float*)take((size_t)E * sizeof(float));
  float* deg   = (float*)take((size_t)N * sizeof(float));
  float* nrm   = (float*)take((size_t)N * sizeof(float));
  float* invdt = (float*)take((size_t)N * sizeof(float));

  const int BLK = 256;
  auto nblk = [](long n, int b) { return (int)((n + b - 1) / b); };

  // 1) X = feat @ W_bef + b_bef ; Y = X (fused second store)
  {
    long threads = (long)(N >> 4) * 4 * 32;
    wmma_gemm64<128, false><<<nblk(threads, BLK), BLK, 0, stream>>>(
        feat, W_bef, b_bef, X, Y, N);
  }

  // 2) w = 1, deg = in-degree, node coefficients
  zero_f<<<nblk(N, BLK), BLK, 0, stream>>>(deg, N);
  init_edges<<<nblk(E, BLK), BLK, 0, stream>>>(dst, w, deg, E);
  node_coeffs<<<nblk(N, BLK), BLK, 0, stream>>>(deg, nrm, invdt, N);

  // 3) 8 propagation steps, attention recompute after step 4 (k==3)
  for (int k = 0; k < 8; ++k) {
    zero_f<<<nblk((long)N * 64, BLK), BLK, 0, stream>>>(P, (long)N * 64);
    edge_scatter<<<nblk((long)E * 16, BLK), BLK, 0, stream>>>(
        src, dst, w, nrm, Y, P, E);
    node_update<<<nblk((long)N * 16, BLK), BLK, 0, stream>>>(
        X, P, nrm, invdt, Y, N);
    if (k == 3) {
      zero_f<<<nblk(N, BLK), BLK, 0, stream>>>(deg, N);
      edge_attention<<<nblk(E, BLK), BLK, 0, stream>>>(src, dst, Y, w, deg, E);
      node_coeffs<<<nblk(N, BLK), BLK, 0, stream>>>(deg, nrm, invdt, N);
    }
  }

  // 4) out = relu(Y) @ W_aft + b_aft
  {
    long threads = (long)(N >> 4) * 4 * 32;
    wmma_gemm64<64, true><<<nblk(threads, BLK), BLK, 0, stream>>>(
        Y, W_aft, b_aft, (float*)d_out, nullptr, N);
  }
}